// DbeCom_50233937494401
// MI455X (gfx1250) — compile-verified
//
#include <hip/hip_runtime.h>
#include <stdint.h>

// Problem shape from the reference setup: (1, 1, 2048, 4096), fp32 binary maps.
#define H_DIM 2048
#define W_DIM 4096
#define THRF 10.0f

// d_ws layout:
//   bytes [0..15]  : accumulators {float num, uint filt, uint npred, uint ngt}
//   bytes [64.. )  : uint8 codes[H*W]
//     low nibble  = |dy| to nearest gt   fg within +-10 vertically (15 = none)
//     high nibble = |dy| to nearest pred fg within +-10 vertically (15 = none)

__global__ void dbe_init(unsigned* acc) {
  if (threadIdx.x < 4) acc[threadIdx.x] = 0u;
}

// ---------------------------------------------------------------------------
// Pass A: vertical min-conv. Each thread owns one column x and a strip of 64
// output rows. It builds an 84-bit foreground mask per array (rows ys-10 ..
// ys+73, two u64 words) and extracts nearest-set-bit distances with ctz/clz.
// Halo amplification 84/64 = 1.31x -> ~84 MB read; writes 1 packed byte per
// pixel (8 MB total).
// ---------------------------------------------------------------------------
__global__ __launch_bounds__(256) void dbe_vpass(
    const float* __restrict__ gt, const float* __restrict__ pred,
    uint8_t* __restrict__ codes) {
  const int gx    = blockIdx.x & 15;   // W/256 = 16 column groups
  const int strip = blockIdx.x >> 4;   // H/64  = 32 row strips
  const int x  = gx * 256 + threadIdx.x;
  const int ys = strip * 64;

  // bit j (j = 0..83) <-> row ys-10+j ; lo holds bits 0..63, hi bits 64..83
  unsigned long long glo = 0ull, ghi = 0ull, plo = 0ull, phi = 0ull;
  const int r0 = ys - 10;
  for (int j = 0; j < 84; ++j) {
    const int r = r0 + j;
    if (r >= 0 && r < H_DIM) {
      const long idx = (long)r * W_DIM + x;
      const unsigned long long bg = (unsigned long long)(gt[idx]   > 0.5f);
      const unsigned long long bp = (unsigned long long)(pred[idx] > 0.5f);
      if (j < 64) { glo |= bg << j;        plo |= bp << j; }
      else        { ghi |= bg << (j - 64); phi |= bp << (j - 64); }
    }
  }

  for (int i = 0; i < 64; ++i) {
    // window for output row ys+i: bits i..i+20, center bit i+10 (dy = bit-10)
    unsigned long long wgl = glo >> i;
    unsigned long long wpl = plo >> i;
    if (i) {                       // bits 64.. feed windows with i >= 44
      wgl |= ghi << (64 - i);
      wpl |= phi << (64 - i);
    }
    const unsigned wg = (unsigned)(wgl & 0x1FFFFFull);
    const unsigned wp = (unsigned)(wpl & 0x1FFFFFull);
    int dg = 15, dp = 15;
    if (wg) {
      const unsigned up = wg >> 10, dn = wg & 0x3FFu;
      const int du = up ? __builtin_ctz(up) : 64;
      const int dd = dn ? (10 - (31 - __builtin_clz(dn))) : 64;
      dg = du < dd ? du : dd;
    }
    if (wp) {
      const unsigned up = wp >> 10, dn = wp & 0x3FFu;
      const int du = up ? __builtin_ctz(up) : 64;
      const int dd = dn ? (10 - (31 - __builtin_clz(dn))) : 64;
      dp = du < dd ? du : dd;
    }
    codes[(long)(ys + i) * W_DIM + x] = (uint8_t)(dg | (dp << 4));
  }
}

// ---------------------------------------------------------------------------
// Pass B: one block per image row. The packed code row (4096 B) is staged
// into LDS with CDNA5 async global->LDS B128 transfers (one per thread),
// fenced by s_wait_asynccnt before the workgroup barrier. Each thread then
// computes the horizontal 21-tap min-conv for 16 pixels and fuses the four
// global reductions (wave32 shfl_xor tree + one atomic per wave).
// ---------------------------------------------------------------------------
__global__ __launch_bounds__(256) void dbe_hpass(
    const uint8_t* __restrict__ codes, float* __restrict__ acc) {
  __shared__ unsigned char tile[16 + W_DIM + 16];
  const int row = blockIdx.x;
  const int tid = threadIdx.x;

  // Halo = "no foreground" (both nibbles 15) for x<0 / x>=W.
  if (tid < 16)       tile[tid]         = 0xFFu;
  else if (tid < 32)  tile[W_DIM + tid] = 0xFFu;   // covers [16+W, 16+W+16)

  // Async stage: each lane copies 16 B from codes[row*W + tid*16] into tile[16 + tid*16].
  const uint8_t* rowp = codes + (size_t)row * W_DIM;
  const unsigned ldsoff = (unsigned)(uintptr_t)(const void*)&tile[16 + tid * 16];
  const unsigned voff   = (unsigned)(tid * 16);
  asm volatile("global_load_async_to_lds_b128 %0, %1, %2"
               :: "v"(ldsoff), "v"(voff), "s"(rowp)
               : "memory");
  asm volatile("s_wait_asynccnt 0" ::: "memory");
  __syncthreads();

  float num = 0.0f;
  unsigned filt = 0u, np = 0u, ng = 0u;
  const int xbase = tid * 16;
  for (int px = 0; px < 16; ++px) {
    const int base = 6 + xbase + px;   // = 16 + (xbase+px) - 10
    int d2g = 4096, d2p = 4096;
#pragma unroll
    for (int dx = 0; dx < 21; ++dx) {
      const int c  = tile[base + dx];
      const int w2 = (dx - 10) * (dx - 10);
      const int a = c & 15, b = c >> 4;
      const int vg = a * a + w2;       // "none"=15 -> >=225 -> clamps, exact below 100
      const int vp = b * b + w2;
      d2g = vg < d2g ? vg : d2g;
      d2p = vp < d2p ? vp : d2p;
    }
    const int ctr = tile[base + 10];
    const int isg = ((ctr & 15) == 0);   // gt pixel  <=> vertical dist 0
    const int isp = ((ctr >> 4) == 0);   // pred pixel
    const float Dg = fminf(sqrtf((float)d2g), THRF);  // D_gt  (dist to gt)
    const float Dp = fminf(sqrtf((float)d2p), THRF);  // D_est (dist to pred)
    // ch1 = min(D_gt*pred, thr); ch2 = min(D_est*gt, thr); pred,gt in {0,1}
    num  += (isp ? Dg : 0.0f) + (isg ? Dp : 0.0f);
    filt += (unsigned)(isp & (d2g < 100));            // pred & (D_gt < thr)
    np   += (unsigned)isp;
    ng   += (unsigned)isg;
  }

  // wave32 reduction (warpSize == 32 on gfx1250)
  for (int off = 16; off > 0; off >>= 1) {
    num  += __shfl_xor(num,  off, 32);
    filt += __shfl_xor(filt, off, 32);
    np   += __shfl_xor(np,   off, 32);
    ng   += __shfl_xor(ng,   off, 32);
  }
  if ((tid & 31) == 0) {
    atomicAdd(&acc[0], num);
    atomicAdd((unsigned*)acc + 1, filt);
    atomicAdd((unsigned*)acc + 2, np);
    atomicAdd((unsigned*)acc + 3, ng);
  }
}

__global__ void dbe_final(const float* acc, float* out) {
  if (threadIdx.x == 0) {
    const float    num  = acc[0];
    const unsigned filt = ((const unsigned*)acc)[1];
    const unsigned np   = ((const unsigned*)acc)[2];
    const unsigned ng   = ((const unsigned*)acc)[3];
    const float den = (float)np + (float)ng;   // B=1: mean == the single sum
    out[0] = (filt == 0u) ? THRF : num / den;
  }
}

extern "C" void kernel_launch(void* const* d_in, const int* in_sizes, int n_in,
                              void* d_out, int out_size, void* d_ws, size_t ws_size,
                              hipStream_t stream) {
  (void)in_sizes; (void)n_in; (void)out_size; (void)ws_size;
  const float* gt   = (const float*)d_in[0];
  const float* pred = (const float*)d_in[1];
  float*   acc   = (float*)d_ws;
  uint8_t* codes = (uint8_t*)d_ws + 64;

  dbe_init <<<1, 64, 0, stream>>>((unsigned*)acc);
  dbe_vpass<<<(W_DIM / 256) * (H_DIM / 64), 256, 0, stream>>>(gt, pred, codes);
  dbe_hpass<<<H_DIM, 256, 0, stream>>>(codes, acc);
  dbe_final<<<1, 32, 0, stream>>>(acc, (float*)d_out);
}